// DistanceAwareSegmentationLoss_72679436583408
// MI455X (gfx1250) — compile-verified
//
#include <hip/hip_runtime.h>
#include <stdint.h>

// ---------------- problem constants (fixed shapes from reference) ----------
#define HGT   512
#define WID   512
#define NBAT  8
#define TILE  32
#define RAD   4                 // BW-1
#define MHALO 5                 // mask halo (edges need one extra row/col up-left)
#define MW    (TILE + 2*MHALO)  // 42  mask tile width
#define EW    (TILE + 2*RAD)    // 40  edge tile width
#define NB_X  (WID/TILE)        // 16
#define NB_Y  (HGT/TILE)        // 16
#define NPART (NBAT*NB_X*NB_Y)  // 2048 per-block partial rows
#define NCOMP 7                 // wce, wsum, p1, t1, p1t1, bcnt, bcorrect
#define NPIX  ((double)(NBAT)*(double)HGT*(double)WID)

struct Smem {
  int           mask[MW*MW];      // 42x42 class ids (clamped halo)
  unsigned char edge[EW*EW];      // 40x40 edge indicator
  unsigned char m1[EW*TILE];      // per edge-row min |ox| (0..4, 7=none), 4B-aligned rows
  float         red[8][NCOMP];    // per-wave partials
};

__global__ __launch_bounds__(256)
void seg_loss_stage1(const float* __restrict__ pred,
                     const int*  __restrict__ tgt,
                     float*      __restrict__ partials) {
  __shared__ Smem sm;
  const int tid = threadIdx.x;
  const int bx = blockIdx.x, by = blockIdx.y, b = blockIdx.z;
  const int x0 = bx * TILE, y0 = by * TILE;
  const int* timg = tgt + (size_t)b * HGT * WID;

  // Main-pass ownership: one tile row x 4 consecutive columns per thread.
  const int row = tid >> 3;          // 0..31
  const int col = (tid & 7) * 4;     // 0,4,...,28
  const int gy  = y0 + row;
  const int gx  = x0 + col;
  const size_t pbase = (size_t)b * 3 * HGT * WID;
  const size_t poff  = pbase + (size_t)gy * WID + gx;   // 16B aligned

  // ---- 1) async global->LDS copy of clamped 42x42 target halo tile -------
  // CDNA5 async path: GLOBAL_LOAD_ASYNC_TO_LDS_B32 (GVS: saddr + vaddr32),
  // tracked by ASYNCcnt; LDS dest address comes from the VDST VGPR.
  for (int t = tid; t < MW*MW; t += 256) {
    int r = t / MW, c = t - r * MW;
    int cy = y0 - MHALO + r; cy = cy < 0 ? 0 : (cy > HGT-1 ? HGT-1 : cy);
    int cx = x0 - MHALO + c; cx = cx < 0 ? 0 : (cx > WID-1 ? WID-1 : cx);
    unsigned srcOff  = (unsigned)((cy * WID + cx) * 4);
    unsigned ldsAddr = (unsigned)(uintptr_t)(&sm.mask[t]); // low 32b of flat LDS addr
    asm volatile("global_load_async_to_lds_b32 %0, %1, %2"
                 :: "v"(ldsAddr), "v"(srcOff), "s"(timg) : "memory");
  }

  // ---- 1b) prefetch the prediction cachelines while the async copy and the
  //          LDS edge/m1 passes run (global_prefetch_b8, gfx1250 path) ------
  __builtin_prefetch(pred + poff, 0, 0);
  __builtin_prefetch(pred + poff + (size_t)HGT * WID, 0, 0);
  __builtin_prefetch(pred + poff + (size_t)2 * HGT * WID, 0, 0);

  asm volatile("s_wait_asynccnt 0" ::: "memory");
  __syncthreads();

  // ---- 2) 40x40 edge map (coordinate-guarded; OOB edges are False) -------
  for (int t = tid; t < EW*EW; t += 256) {
    int r = t / EW, c = t - r * EW;
    int ey = y0 - RAD + r, ex = x0 - RAD + c;
    unsigned char e = 0;
    if (ey >= 0 && ey < HGT && ex >= 0 && ex < WID) {
      int m = sm.mask[(r+1)*MW + (c+1)];
      if (ey > 0 && m != sm.mask[r*MW + (c+1)]) e = 1;       // dy (prepend dup)
      if (ex > 0 && m != sm.mask[(r+1)*MW + c]) e = 1;       // dx
    }
    sm.edge[t] = e;
  }
  __syncthreads();

  // ---- 3a) separable pass 1: per edge-row min |ox| within +-4 ------------
  for (int t = tid; t < EW*TILE; t += 256) {
    int r = t / TILE, j = t - r * TILE;           // j = output column
    const unsigned char* er = &sm.edge[r*EW + j]; // covers ox = -4..+4
    int best = 7;
    #pragma unroll
    for (int k = 0; k <= 2*RAD; ++k) {
      int a = k - RAD; a = a < 0 ? -a : a;        // |ox|
      if (er[k] && a < best) best = a;
    }
    sm.m1[t] = (unsigned char)best;
  }
  __syncthreads();

  // ---- 3b/4) per-pixel: min d^2, weight, softmax, CE, dice terms ---------
  // 4 pixels per thread, separable pass 2 reads one packed uchar4 per oy.
  int mind2[4] = {25, 25, 25, 25};                 // BW*BW cap
  #pragma unroll
  for (int oy = -RAD; oy <= RAD; ++oy) {
    unsigned mrow = *(const unsigned*)&sm.m1[(row + RAD + oy) * TILE + col];
    #pragma unroll
    for (int k = 0; k < 4; ++k) {
      int v  = (mrow >> (8 * k)) & 0xFF;
      int d2 = oy * oy + v * v;
      if (d2 < mind2[k]) mind2[k] = d2;
    }
  }

  const float4 a4 = *(const float4*)(pred + poff);
  const float4 b4 = *(const float4*)(pred + poff + (size_t)HGT * WID);
  const float4 c4 = *(const float4*)(pred + poff + (size_t)2 * HGT * WID);
  const float xa[4] = {a4.x, a4.y, a4.z, a4.w};
  const float xb[4] = {b4.x, b4.y, b4.z, b4.w};
  const float xc[4] = {c4.x, c4.y, c4.z, c4.w};

  float wce = 0.f, wsum = 0.f, p1s = 0.f, t1s = 0.f, p1t1 = 0.f;
  float bcnt = 0.f, bcor = 0.f;

  #pragma unroll
  for (int k = 0; k < 4; ++k) {
    float w = 1.0f;
    if (mind2[k] < 25) w = 2.0f - sqrtf((float)mind2[k]) * 0.2f; // 1+(2-1)*(1-d/5)

    const int tc = sm.mask[(row + MHALO) * MW + (col + k + MHALO)];

    const float mx = fmaxf(xa[k], fmaxf(xb[k], xc[k]));
    const float ea = expf(xa[k] - mx), eb = expf(xb[k] - mx), ec = expf(xc[k] - mx);
    const float s  = ea + eb + ec;
    const float lse = mx + logf(s);
    const float xt = (tc == 0) ? xa[k] : ((tc == 1) ? xb[k] : xc[k]);
    const float ce = lse - xt;
    const float p1 = eb / s;

    int pc = 0; float bst = xa[k];                 // argmax, first index on ties
    if (xb[k] > bst) { bst = xb[k]; pc = 1; }
    if (xc[k] > bst) { pc = 2; }

    wce  += ce * w;
    wsum += w;
    p1s  += p1;
    const float t1 = (tc == 1) ? 1.0f : 0.0f;
    t1s  += t1;
    p1t1 += p1 * t1;
    if (w > 1.5f) { bcnt += 1.0f; if (pc == tc) bcor += 1.0f; }
  }

  // ---- 5) wave32 shuffle reduction -> per-wave LDS -> per-block partial --
  const int lane = tid & 31, wv = tid >> 5;
  float vals[NCOMP] = {wce, wsum, p1s, t1s, p1t1, bcnt, bcor};
  #pragma unroll
  for (int c = 0; c < NCOMP; ++c) {
    float v = vals[c];
    for (int d = 16; d > 0; d >>= 1) v += __shfl_down(v, d, 32);
    if (lane == 0) sm.red[wv][c] = v;
  }
  __syncthreads();
  if (tid == 0) {
    const int blk = (b * NB_Y + by) * NB_X + bx;
    #pragma unroll
    for (int c = 0; c < NCOMP; ++c) {
      float a = 0.f;
      #pragma unroll
      for (int w8 = 0; w8 < 8; ++w8) a += sm.red[w8][c];
      partials[blk * NCOMP + c] = a;
    }
  }
}

__global__ __launch_bounds__(256)
void seg_loss_stage2(const float* __restrict__ partials, float* __restrict__ out) {
  __shared__ double red[8][NCOMP];
  double acc[NCOMP] = {0, 0, 0, 0, 0, 0, 0};
  for (int p = threadIdx.x; p < NPART; p += 256) {
    #pragma unroll
    for (int c = 0; c < NCOMP; ++c) acc[c] += (double)partials[p * NCOMP + c];
  }
  const int lane = threadIdx.x & 31, wv = threadIdx.x >> 5;
  #pragma unroll
  for (int c = 0; c < NCOMP; ++c) {
    double v = acc[c];
    for (int d = 16; d > 0; d >>= 1) v += __shfl_down(v, d, 32);
    if (lane == 0) red[wv][c] = v;
  }
  __syncthreads();
  if (threadIdx.x == 0) {
    double s[NCOMP];
    #pragma unroll
    for (int c = 0; c < NCOMP; ++c) {
      double a = 0.0;
      #pragma unroll
      for (int w = 0; w < 8; ++w) a += red[w][c];
      s[c] = a;
    }
    const double invN = 1.0 / NPIX;
    const double wce  = s[0] * invN;
    const double avgw = s[1] * invN;
    const double dice = (2.0 * s[4] + 1.0) / (s[2] + s[3] + 1.0);
    const double dl   = 1.0 - dice;
    const double tot  = wce + dl;
    const double bacc = (s[5] > 0.0) ? (s[6] / s[5]) : 0.0;
    out[0] = (float)tot;   // total_loss
    out[1] = (float)wce;   // weighted_ce
    out[2] = (float)dl;    // dice_loss
    out[3] = (float)bacc;  // boundary_acc
    out[4] = (float)avgw;  // avg_weight
  }
}

extern "C" void kernel_launch(void* const* d_in, const int* in_sizes, int n_in,
                              void* d_out, int out_size, void* d_ws, size_t ws_size,
                              hipStream_t stream) {
  const float* pred = (const float*)d_in[0];
  const int*   tgt  = (const int*)d_in[1];
  float* partials   = (float*)d_ws;              // NPART*NCOMP floats (~57 KB)

  dim3 grid(NB_X, NB_Y, NBAT);
  seg_loss_stage1<<<grid, 256, 0, stream>>>(pred, tgt, partials);
  seg_loss_stage2<<<1, 256, 0, stream>>>(partials, (float*)d_out);
}